// LearnedCacheKVLayer_57226144252196
// MI455X (gfx1250) — compile-verified
//
#include <hip/hip_runtime.h>
#include <stdint.h>

// Shapes fixed by the reference.
#define B_     2
#define S_     4096
#define H_     16
#define D_     128
#define MAXSEQ 4096
#define HD     (H_ * D_)     // 2048 floats per (b,s) row  = 8 KiB
#define ROW4   (HD / 4)      // 512 float4 per row
#define NT     256           // threads per block (8 wave32)

typedef float v4f __attribute__((ext_vector_type(4)));

// ---- CDNA5 async global<->LDS helpers (ASYNCcnt-tracked, per CDNA5 ISA) ----
__device__ __forceinline__ uint32_t lds_off(const void* p) {
  // Flat LDS address: low 32 bits == LDS byte offset (aperture rule, ISA 10.2)
  return (uint32_t)(uintptr_t)p;
}
// RT load: source has cross-kernel reuse (k/v rows re-read by gather pass) -> let L2 keep it
__device__ __forceinline__ void async_load_b128(uint32_t lds_dst, const void* g) {
  asm volatile("global_load_async_to_lds_b128 %0, %1, off"
               :: "v"(lds_dst), "v"((uint64_t)(uintptr_t)g)
               : "memory");
}
// NT load: read-once stream (old cache rows) -> don't pollute L2
__device__ __forceinline__ void async_load_b128_nt(uint32_t lds_dst, const void* g) {
  asm volatile("global_load_async_to_lds_b128 %0, %1, off th:TH_LOAD_NT"
               :: "v"(lds_dst), "v"((uint64_t)(uintptr_t)g)
               : "memory");
}
// NT store: write-once output stream
__device__ __forceinline__ void async_store_b128_nt(void* g, uint32_t lds_src) {
  asm volatile("global_store_async_from_lds_b128 %0, %1, off th:TH_STORE_NT"
               :: "v"((uint64_t)(uintptr_t)g), "v"(lds_src)
               : "memory");
}
__device__ __forceinline__ void wait_async0() {
  asm volatile("s_wait_asynccnt 0x0" ::: "memory");
}

// ---- Pass 1: inverse scatter map. target[b*MAXSEQ + pos] = s (distinct pos -> race-free) ----
__global__ void build_target_kernel(const int* __restrict__ pos,
                                    const int* __restrict__ mask,
                                    int* __restrict__ target) {
  int i = blockIdx.x * blockDim.x + threadIdx.x;
  if (i >= B_ * S_) return;
  int b = i / S_;
  int p = pos[i];
  if (mask[i] != 0 && p < MAXSEQ) target[b * MAXSEQ + p] = i - b * S_;   // s index
}

// ---- Pass 2: rebuild cache. One block per cache row; async LDS staging. ----
__global__ __launch_bounds__(NT)
void rebuild_cache_kernel(const float* __restrict__ k,  const float* __restrict__ v,
                          const float* __restrict__ ck, const float* __restrict__ cv,
                          const int*   __restrict__ target,
                          float* __restrict__ nck, float* __restrict__ ncv) {
  __shared__ v4f bk[ROW4];   // 8 KiB
  __shared__ v4f bv[ROW4];   // 8 KiB
  int row = blockIdx.x;                       // 0 .. B*MAXSEQ-1
  int b   = row / MAXSEQ;
  int t   = target[row];
  bool fresh = (t >= 0);

  if (fresh) {
    // Fresh k/v rows: these same rows are re-read by the gather pass -> RT (cacheable).
    size_t src_row = (size_t)b * S_ + (size_t)t;
    const v4f* sk = (const v4f*)(k + src_row * HD);
    const v4f* sv = (const v4f*)(v + src_row * HD);
    for (int i = threadIdx.x; i < ROW4; i += NT) {
      async_load_b128(lds_off(&bk[i]), sk + i);
      async_load_b128(lds_off(&bv[i]), sv + i);
    }
  } else {
    // Old cache rows: read exactly once across the whole call -> NT.
    size_t src_row = (size_t)row;
    const v4f* sk = (const v4f*)(ck + src_row * HD);
    const v4f* sv = (const v4f*)(cv + src_row * HD);
    for (int i = threadIdx.x; i < ROW4; i += NT) {
      async_load_b128_nt(lds_off(&bk[i]), sk + i);
      async_load_b128_nt(lds_off(&bv[i]), sv + i);
    }
  }
  wait_async0();                              // loads complete: LDS data valid
  // Each thread owns LDS slots {tid, tid+256}: per-wave-private -> no barrier needed.
  v4f* dk = (v4f*)(nck + (size_t)row * HD);
  v4f* dv = (v4f*)(ncv + (size_t)row * HD);
  for (int i = threadIdx.x; i < ROW4; i += NT) {
    async_store_b128_nt(dk + i, lds_off(&bk[i]));
    async_store_b128_nt(dv + i, lds_off(&bv[i]));
  }
  wait_async0();                              // stores drained before wave exit
}

// ---- Pass 3: gathered k_out / v_out. NT B128 loads, NT B128 stores. ----
__global__ __launch_bounds__(NT)
void gather_out_kernel(const float* __restrict__ k,  const float* __restrict__ v,
                       const float* __restrict__ ck, const float* __restrict__ cv,
                       const int* __restrict__ pos,  const int* __restrict__ mask,
                       const int* __restrict__ cvl_p,
                       float* __restrict__ kout, float* __restrict__ vout) {
  int row = blockIdx.x;                       // 0 .. B*S-1
  int b   = row / S_;
  int p   = pos[row];
  int cvl = cvl_p[0];
  bool gm = (p < cvl) && (mask[row] == 0);    // gather from cache?
  size_t src_row = gm ? ((size_t)b * MAXSEQ + (size_t)p) : (size_t)row;
  const v4f* sk = (const v4f*)((gm ? ck : k) + src_row * HD);
  const v4f* sv = (const v4f*)((gm ? cv : v) + src_row * HD);
  v4f* dk = (v4f*)(kout + (size_t)row * HD);
  v4f* dv = (v4f*)(vout + (size_t)row * HD);
  for (int i = threadIdx.x; i < ROW4; i += NT) {
    // Last consumer of these lines -> NT loads (last-use), NT stores (write-once).
    v4f a = __builtin_nontemporal_load(sk + i);
    v4f c = __builtin_nontemporal_load(sv + i);
    __builtin_nontemporal_store(a, dk + i);
    __builtin_nontemporal_store(c, dv + i);
  }
}

// ---- Pass 4: scalar metrics (hit_rate, new_valid_length, num_updates) ----
__global__ __launch_bounds__(256)
void metrics_kernel(const int* __restrict__ pos, const int* __restrict__ mask,
                    const int* __restrict__ cvl_p, float* __restrict__ out3) {
  __shared__ int sh[256], su[256], sm[256];
  int t = threadIdx.x;
  int cvl = cvl_p[0];
  int hits = 0, upd = 0, mx = -2147483647;
  for (int i = t; i < B_ * S_; i += 256) {
    int p = pos[i];
    hits += (p < cvl) ? 1 : 0;
    upd  += (mask[i] != 0) ? 1 : 0;
    mx    = (p > mx) ? p : mx;
  }
  sh[t] = hits; su[t] = upd; sm[t] = mx;
  __syncthreads();
  for (int off = 128; off > 0; off >>= 1) {
    if (t < off) {
      sh[t] += sh[t + off];
      su[t] += su[t + off];
      sm[t]  = sm[t] > sm[t + off] ? sm[t] : sm[t + off];
    }
    __syncthreads();
  }
  if (t == 0) {
    float eh = 0.01f * (float)sh[0];
    float em = 0.01f * ((float)(B_ * S_) - (float)sh[0]);
    out3[0] = eh / (eh + em + 1e-8f);                       // hit_rate
    int grown = (cvl > sm[0] + 1) ? cvl : sm[0] + 1;
    int nvl = (su[0] > 0) ? (grown < MAXSEQ ? grown : MAXSEQ) : cvl;
    out3[1] = (float)nvl;                                   // new_valid_length
    out3[2] = (float)su[0];                                 // num_updates
  }
}

extern "C" void kernel_launch(void* const* d_in, const int* in_sizes, int n_in,
                              void* d_out, int out_size, void* d_ws, size_t ws_size,
                              hipStream_t stream) {
  (void)in_sizes; (void)n_in; (void)out_size; (void)ws_size;
  const float* k   = (const float*)d_in[0];
  const float* v   = (const float*)d_in[1];
  const float* ck  = (const float*)d_in[2];
  const float* cv  = (const float*)d_in[3];
  const int*   pos = (const int*)d_in[4];
  const int*   msk = (const int*)d_in[5];
  const int*   cvl = (const int*)d_in[6];

  float* o = (float*)d_out;
  const size_t N = (size_t)B_ * S_ * H_ * D_;   // 16,777,216
  float* kout = o;
  float* vout = o + N;
  float* nck  = o + 2 * N;
  float* ncv  = o + 3 * N;
  float* met  = o + 4 * N;

  int* target = (int*)d_ws;                     // B*MAXSEQ ints = 32 KiB
  hipMemsetAsync(target, 0xFF, (size_t)B_ * MAXSEQ * sizeof(int), stream);  // -1

  build_target_kernel<<<(B_ * S_ + 255) / 256, 256, 0, stream>>>(pos, msk, target);
  rebuild_cache_kernel<<<B_ * MAXSEQ, NT, 0, stream>>>(k, v, ck, cv, target, nck, ncv);
  gather_out_kernel<<<B_ * S_, NT, 0, stream>>>(k, v, ck, cv, pos, msk, cvl, kout, vout);
  metrics_kernel<<<1, 256, 0, stream>>>(pos, msk, cvl, met);
}